// GNN_with_params_73727408603846
// MI455X (gfx1250) — compile-verified
//
#include <hip/hip_runtime.h>
#include <stdint.h>

#define D 256
#define EPS 1e-5f

typedef __attribute__((ext_vector_type(2))) float v2f;
typedef __attribute__((ext_vector_type(8))) float v8f;

// ---------------- small helpers ----------------

__global__ void zero_f32(float* __restrict__ p, int n) {
    int i = blockIdx.x * blockDim.x + threadIdx.x;
    if (i < n) p[i] = 0.0f;
}

// deg[c] += 1 for every edge target (E graph edges + N self loops)
__global__ void degree_kernel(const int* __restrict__ col, float* deg, int E, int N) {
    int i = blockIdx.x * blockDim.x + threadIdx.x;
    int total = E + N;
    if (i >= total) return;
    int c = (i < E) ? col[i] : (i - E);
    atomicAdd(&deg[c], 1.0f);
}

// deg -> deg^-0.5 in place (inf -> 0 as in reference)
__global__ void invsqrt_kernel(float* __restrict__ deg, int n) {
    int i = blockIdx.x * blockDim.x + threadIdx.x;
    if (i >= n) return;
    float d = deg[i];
    deg[i] = (d > 0.0f) ? rsqrtf(d) : 0.0f;
}

// out[i*D+d] = bias[d]  (aggregation buffer starts at +b)
__global__ void init_bias(float* __restrict__ out, const float* __restrict__ b, size_t total) {
    size_t i = (size_t)blockIdx.x * blockDim.x + threadIdx.x;
    if (i < total) out[i] = b[i & (D - 1)];
}

// ---------------- fp32 WMMA GEMM: H[N,D] = X[N,D] @ W[D,D] ----------------
// One block = 8 waves, one 16-row stripe of X staged in LDS.
// Each wave owns two 16x16 output tiles (cols wave*16 and (wave+8)*16),
// reusing its A fragment across both V_WMMA_F32_16X16X4_F32 issues.
__global__ __launch_bounds__(256)
void gemm_wmma_f32(const float* __restrict__ X, const float* __restrict__ W,
                   float* __restrict__ H, int N) {
    __shared__ float xs[16 * D];                 // 16 KB A tile
    const int tid = threadIdx.x;
    const int rowBase = blockIdx.x * 16;

    // cooperative, coalesced load of the 16 x 256 A tile (zero-pad tail rows)
    for (int idx = tid; idx < 16 * D; idx += 256) {
        int r = idx >> 8;
        int c = idx & 255;
        int gr = rowBase + r;
        xs[idx] = (gr < N) ? X[(size_t)gr * D + c] : 0.0f;
    }
    __syncthreads();

    const int lane = tid & 31;
    const int wave = tid >> 5;
    const int m    = lane & 15;     // A row within tile / B,N column within tile
    const int hi   = lane >> 4;     // half-wave select
    const int koff = hi * 2;        // K pairs: lanes 0-15 -> K,K+1 ; lanes 16-31 -> K+2,K+3
    const int col0 = wave * 16;
    const int col1 = (wave + 8) * 16;

    v8f acc0 = {};
    v8f acc1 = {};

    for (int kk = 0; kk < D; kk += 4) {
        v2f a, b0, b1;
        a.x = xs[m * D + kk + koff];
        a.y = xs[m * D + kk + koff + 1];
        const float* wr0 = W + (size_t)(kk + koff) * D;
        const float* wr1 = W + (size_t)(kk + koff + 1) * D;
        b0.x = wr0[col0 + m];
        b0.y = wr1[col0 + m];
        b1.x = wr0[col1 + m];
        b1.y = wr1[col1 + m];
        // (neg_a, A, neg_b, B, c_mod, C, reuse_a, reuse_b)
        acc0 = __builtin_amdgcn_wmma_f32_16x16x4_f32(false, a, false, b0,
                                                     (short)0, acc0, false, false);
        acc1 = __builtin_amdgcn_wmma_f32_16x16x4_f32(false, a, false, b1,
                                                     (short)0, acc1, false, false);
    }

    // C/D layout: VGPR r -> M=r (lanes 0-15), M=r+8 (lanes 16-31); N = lane&15
    const int rowOff = rowBase + (hi ? 8 : 0);
    if (rowBase + 16 <= N) {
        // fast path: whole tile in range -> branchless stores with a single
        // precomputed base pointer and constant row-stride offsets.
        float* h0 = H + (size_t)rowOff * D + col0 + m;
        float* h1 = H + (size_t)rowOff * D + col1 + m;
#pragma unroll
        for (int r = 0; r < 8; ++r) {
            h0[(size_t)r * D] = acc0[r];
            h1[(size_t)r * D] = acc1[r];
        }
    } else {
#pragma unroll
        for (int r = 0; r < 8; ++r) {
            int gm = rowOff + r;
            if (gm < N) {
                H[(size_t)gm * D + col0 + m] = acc0[r];
                H[(size_t)gm * D + col1 + m] = acc1[r];
            }
        }
    }
}

// ---------------- edge scatter: agg[col] += h[row] * dis[row]*dis[col] ----------------
// One wave32 per edge; lane-strided so both the gather loads and the float
// atomics are fully coalesced. Self-loops are the last N virtual edges.
__global__ __launch_bounds__(256)
void scatter_edges(const int* __restrict__ row, const int* __restrict__ col,
                   const float* __restrict__ dis, const float* __restrict__ h,
                   float* agg, int E, int N) {
    int gid  = blockIdx.x * blockDim.x + threadIdx.x;
    int e    = gid >> 5;
    int lane = gid & 31;
    int total = E + N;
    if (e >= total) return;
    int r, c;
    if (e < E) { r = row[e]; c = col[e]; }
    else       { r = e - E;  c = r; }
    float norm = dis[r] * dis[c];
    const float* hr = h + (size_t)r * D;
    float* ac = agg + (size_t)c * D;
#pragma unroll
    for (int j = 0; j < 8; ++j) {
        int f = lane + j * 32;
        atomicAdd(&ac[f], hr[f] * norm);
    }
}

// ---------------- BatchNorm (training mode) over relu(z), deterministic ----------------
// pass 1: each of 256 blocks owns a contiguous row chunk; thread d accumulates
// column-d partial sum / sumsq with coalesced reads.
__global__ __launch_bounds__(256)
void bn_stats(const float* __restrict__ z, float* __restrict__ psum,
              float* __restrict__ psq, int N, int rowsPerBlock) {
    int d  = threadIdx.x;
    int r0 = blockIdx.x * rowsPerBlock;
    int r1 = r0 + rowsPerBlock;
    if (r1 > N) r1 = N;
    float s = 0.0f, q = 0.0f;
    for (int r = r0; r < r1; ++r) {
        float v = z[(size_t)r * D + d];
        v = (v > 0.0f) ? v : 0.0f;          // fused ReLU
        s += v;
        q += v * v;
    }
    psum[blockIdx.x * D + d] = s;
    psq [blockIdx.x * D + d] = q;
}

// pass 2: single block folds the 256 partials in fixed order.
__global__ void bn_finalize(const float* __restrict__ psum, const float* __restrict__ psq,
                            float* __restrict__ mean, float* __restrict__ var,
                            int nblocks, int N) {
    int d = threadIdx.x;
    float s = 0.0f, q = 0.0f;
    for (int b = 0; b < nblocks; ++b) {
        s += psum[b * D + d];
        q += psq [b * D + d];
    }
    float mu = s / (float)N;
    mean[d] = mu;
    var[d]  = q / (float)N - mu * mu;       // biased variance
}

__global__ void bn_apply(const float* __restrict__ z, const float* __restrict__ mean,
                         const float* __restrict__ var, const float* __restrict__ g,
                         const float* __restrict__ bt, float* __restrict__ out,
                         size_t total) {
    size_t i = (size_t)blockIdx.x * blockDim.x + threadIdx.x;
    if (i >= total) return;
    int d = (int)(i & (D - 1));
    float v = z[i];
    v = (v > 0.0f) ? v : 0.0f;              // fused ReLU
    out[i] = g[d] * (v - mean[d]) * rsqrtf(var[d] + EPS) + bt[d];
}

// ---------------- driver ----------------

extern "C" void kernel_launch(void* const* d_in, const int* in_sizes, int n_in,
                              void* d_out, int out_size, void* d_ws, size_t ws_size,
                              hipStream_t stream) {
    const float* x   = (const float*)d_in[0];
    const int*   ei  = (const int*)  d_in[1];   // edge_index [2, E], int32 under JAX defaults
    const float* W0  = (const float*)d_in[2];
    const float* b0  = (const float*)d_in[3];
    const float* g0  = (const float*)d_in[4];
    const float* bt0 = (const float*)d_in[5];
    const float* W1  = (const float*)d_in[6];
    const float* b1  = (const float*)d_in[7];
    const float* g1  = (const float*)d_in[8];
    const float* bt1 = (const float*)d_in[9];

    const int N = in_sizes[0] / D;
    const int E = in_sizes[1] / 2;
    const int* row = ei;
    const int* col = ei + E;

    // workspace carve-up (floats)
    float* ws   = (float*)d_ws;
    float* dis  = ws;             ws += N;          // degree -> deg^-0.5
    float* psum = ws;             ws += 256 * D;    // BN partials
    float* psq  = ws;             ws += 256 * D;
    float* mean = ws;             ws += D;
    float* var  = ws;             ws += D;
    ws = (float*)(((uintptr_t)ws + 63) & ~(uintptr_t)63);
    float* bufA = ws;             ws += (size_t)N * D;
    float* bufB = ws;

    const int    EN            = E + N;
    const int    rowsPerBlock  = (N + 255) / 256;
    const size_t total         = (size_t)N * D;
    const int    blocksTotal   = (int)((total + 255) / 256);
    const int    gemmBlocks    = (N + 15) / 16;

    // symmetric normalization coefficients
    zero_f32      <<<(N + 255) / 256, 256, 0, stream>>>(dis, N);
    degree_kernel <<<(EN + 255) / 256, 256, 0, stream>>>(col, dis, E, N);
    invsqrt_kernel<<<(N + 255) / 256, 256, 0, stream>>>(dis, N);

    // ---- layer 0 ----
    gemm_wmma_f32 <<<gemmBlocks, 256, 0, stream>>>(x, W0, bufA, N);
    init_bias     <<<blocksTotal, 256, 0, stream>>>(bufB, b0, total);
    scatter_edges <<<(EN + 7) / 8, 256, 0, stream>>>(row, col, dis, bufA, bufB, E, N);
    bn_stats      <<<256, 256, 0, stream>>>(bufB, psum, psq, N, rowsPerBlock);
    bn_finalize   <<<1, 256, 0, stream>>>(psum, psq, mean, var, 256, N);
    bn_apply      <<<blocksTotal, 256, 0, stream>>>(bufB, mean, var, g0, bt0, bufA, total);

    // ---- layer 1 ----
    gemm_wmma_f32 <<<gemmBlocks, 256, 0, stream>>>(bufA, W1, bufB, N);
    init_bias     <<<blocksTotal, 256, 0, stream>>>(bufA, b1, total);
    scatter_edges <<<(EN + 7) / 8, 256, 0, stream>>>(row, col, dis, bufB, bufA, E, N);
    bn_stats      <<<256, 256, 0, stream>>>(bufA, psum, psq, N, rowsPerBlock);
    bn_finalize   <<<1, 256, 0, stream>>>(psum, psq, mean, var, 256, N);
    bn_apply      <<<blocksTotal, 256, 0, stream>>>(bufA, mean, var, g1, bt1, (float*)d_out, total);
}